// LongRangeDistanceModule_42958262895191
// MI455X (gfx1250) — compile-verified
//
#include <hip/hip_runtime.h>
#include <math.h>

typedef __attribute__((ext_vector_type(2))) float v2f;
typedef __attribute__((ext_vector_type(8))) float v8f;

#define NUM_BINS 64
#define MAX_DISTF 25.0f
#define HIDDEN 256
#define BGRAPH 64
#define SPTS 1024
#define SPLIT 8
#define NWAVES 8

// ---------------------------------------------------------------------------
// ws layout:
//   [0, 16KB)   : unsigned ghist[BGRAPH * NUM_BINS]   (raw counts)
//   [16KB,32KB) : float    nhist[BGRAPH * NUM_BINS]   (normalized histograms)
// ---------------------------------------------------------------------------

__global__ void zero_hist_kernel(unsigned* __restrict__ ghist) {
    int i = blockIdx.x * blockDim.x + threadIdx.x;
    if (i < BGRAPH * NUM_BINS) ghist[i] = 0u;
}

// One graph handled by SPLIT blocks; each block: 8 waves, points + histogram in LDS.
// Each 16x16 (i,j) tile's squared distances come from ONE v_wmma_f32_16x16x4_f32:
//   A row i = (-2x, -2y, -2z, |p_i|^2), B col j = (x, y, z, 1)
//   => D[i][j] = |p_i|^2 - 2 p_i.p_j ; add |p_j|^2 per column afterwards.
__global__ __launch_bounds__(256) void hist_kernel(const float* __restrict__ pos,
                                                   unsigned* __restrict__ ghist) {
    __shared__ float4   pts[SPTS];          // x, y, z, |p|^2   (16 KB)
    __shared__ unsigned histL[NUM_BINS];

    const int b     = blockIdx.x >> 3;          // graph id
    const int split = blockIdx.x & (SPLIT - 1); // which slice of tiles
    const int tid   = threadIdx.x;
    const int w     = tid >> 5;
    const int lane  = tid & 31;
    const int m     = lane & 15;   // row/col within 16-wide tile
    const int half  = lane >> 4;   // 0: K0/K1 & rows 0-7 ; 1: K2/K3 & rows 8-15

    // stage this graph's points into LDS (with precomputed squared norm)
    for (int i = tid; i < SPTS; i += 256) {
        const float* p = pos + ((size_t)b * SPTS + i) * 3;
        float x = p[0], y = p[1], z = p[2];
        pts[i] = make_float4(x, y, z, x * x + y * y + z * z);
    }
    if (tid < NUM_BINS) histL[tid] = 0u;
    __syncthreads();

    // 64x64 tile grid over (i,j); keep only tj >= ti (upper triangle of tiles).
    const int flatWave = split * NWAVES + w;       // 0..63, wave-uniform
    for (int t = flatWave; t < 64 * 64; t += SPLIT * NWAVES) {
        const int ti = t >> 6;
        const int tj = t & 63;
        if (tj < ti) continue;                     // wave-uniform skip

        const float4 p = pts[ti * 16 + m];         // A-matrix row point
        const float4 q = pts[tj * 16 + m];         // B-matrix col point

        // 32-bit A 16x4 layout: lanes0-15 hold K0,K1 ; lanes16-31 hold K2,K3
        v2f a, bb;
        a[0]  = half ? (-2.0f * p.z) : (-2.0f * p.x);
        a[1]  = half ? p.w           : (-2.0f * p.y);
        bb[0] = half ? q.z           : q.x;
        bb[1] = half ? 1.0f          : q.y;

        v8f c = {};
        c = __builtin_amdgcn_wmma_f32_16x16x4_f32(false, a, false, bb,
                                                  (short)0, c, false, false);

        const int jcol = tj * 16 + m;              // D col index == lane&15
        #pragma unroll
        for (int v = 0; v < 8; ++v) {
            const int irow = ti * 16 + v + half * 8;   // D row for VGPR v
            float d2   = fmaxf(c[v] + q.w, 0.0f);      // += |p_j|^2, clamp
            // validity on d^2 (no sqrt dependency); raw v_sqrt_f32 for binning
            bool valid = (irow < jcol) && (d2 <= MAX_DISTF * MAX_DISTF);
            float dist = __builtin_amdgcn_sqrtf(d2);
            int bin = (int)(dist * (NUM_BINS / MAX_DISTF));
            bin = bin > (NUM_BINS - 1) ? (NUM_BINS - 1) : bin;
            if (valid) {
                atomicAdd(&histL[bin], 1u);            // ds_add_u32
            }
        }
    }
    __syncthreads();
    if (tid < NUM_BINS) atomicAdd(&ghist[b * NUM_BINS + tid], histL[tid]);
}

__global__ void norm_kernel(const unsigned* __restrict__ ghist,
                            float* __restrict__ nhist) {
    const int b = blockIdx.x;
    const int t = threadIdx.x;   // 64 threads = 64 bins
    float sum = 0.0f;
    #pragma unroll 8
    for (int i = 0; i < NUM_BINS; ++i) sum += (float)ghist[b * NUM_BINS + i];
    nhist[b * NUM_BINS + t] = (float)ghist[b * NUM_BINS + t] / (sum + 1e-8f);
}

// Fused MLP: out = silu(hist @ W1^T + b1) @ W2^T + b2, all via f32 WMMA.
// One block per 16-row tile (4 blocks); layer-1 result lives in LDS.
__global__ __launch_bounds__(256) void mlp_kernel(const float* __restrict__ nhist,
                                                  const float* __restrict__ W1,
                                                  const float* __restrict__ b1,
                                                  const float* __restrict__ W2,
                                                  const float* __restrict__ b2,
                                                  float* __restrict__ out) {
    __shared__ float hLDS[16 * HIDDEN];    // 16 rows x 256 cols (16 KB)

    const int ti   = blockIdx.x;           // 0..3 row tile
    const int tid  = threadIdx.x;
    const int w    = tid >> 5;
    const int lane = tid & 31;
    const int m    = lane & 15;
    const int half = lane >> 4;
    const int koff = 2 * half;             // lanes0-15: K+0,K+1 ; lanes16-31: K+2,K+3

    // ---- layer 1: h = silu(hist @ W1^T + b1) -> LDS ----
    for (int tj = w; tj < HIDDEN / 16; tj += NWAVES) {
        const int col = tj * 16 + m;
        v8f acc = {};
        for (int k = 0; k < NUM_BINS; k += 4) {
            v2f a, bf;
            a[0]  = nhist[(ti * 16 + m) * NUM_BINS + k + koff];
            a[1]  = nhist[(ti * 16 + m) * NUM_BINS + k + koff + 1];
            bf[0] = W1[col * NUM_BINS + k + koff];      // B[k][n] = W1[n][k]
            bf[1] = W1[col * NUM_BINS + k + koff + 1];
            acc = __builtin_amdgcn_wmma_f32_16x16x4_f32(false, a, false, bf,
                                                        (short)0, acc, false, false);
        }
        const float bias = b1[col];
        #pragma unroll
        for (int v = 0; v < 8; ++v) {
            const int r = v + half * 8;
            float x   = acc[v] + bias;
            float sig = 1.0f / (1.0f + __expf(-x));
            hLDS[r * HIDDEN + col] = x * sig;           // silu
        }
    }
    __syncthreads();

    // ---- layer 2: out = h @ W2^T + b2 ----
    for (int tj = w; tj < HIDDEN / 16; tj += NWAVES) {
        const int col = tj * 16 + m;
        v8f acc = {};
        for (int k = 0; k < HIDDEN; k += 4) {
            v2f a, bf;
            a[0]  = hLDS[m * HIDDEN + k + koff];
            a[1]  = hLDS[m * HIDDEN + k + koff + 1];
            bf[0] = W2[col * HIDDEN + k + koff];        // B[k][n] = W2[n][k]
            bf[1] = W2[col * HIDDEN + k + koff + 1];
            acc = __builtin_amdgcn_wmma_f32_16x16x4_f32(false, a, false, bf,
                                                        (short)0, acc, false, false);
        }
        const float bias = b2[col];
        #pragma unroll
        for (int v = 0; v < 8; ++v) {
            const int r = v + half * 8;
            out[(ti * 16 + r) * HIDDEN + col] = acc[v] + bias;
        }
    }
}

extern "C" void kernel_launch(void* const* d_in, const int* in_sizes, int n_in,
                              void* d_out, int out_size, void* d_ws, size_t ws_size,
                              hipStream_t stream) {
    (void)in_sizes; (void)n_in; (void)out_size; (void)ws_size;
    const float* pos = (const float*)d_in[0];
    const float* W1  = (const float*)d_in[1];
    const float* b1  = (const float*)d_in[2];
    const float* W2  = (const float*)d_in[3];
    const float* b2  = (const float*)d_in[4];
    // d_in[5] = batch ids: unused (contiguous equal segments by construction)
    float* out = (float*)d_out;

    unsigned* ghist = (unsigned*)d_ws;
    float*    nhist = (float*)((char*)d_ws + BGRAPH * NUM_BINS * sizeof(unsigned));

    zero_hist_kernel<<<(BGRAPH * NUM_BINS + 255) / 256, 256, 0, stream>>>(ghist);
    hist_kernel<<<BGRAPH * SPLIT, 256, 0, stream>>>(pos, ghist);
    norm_kernel<<<BGRAPH, NUM_BINS, 0, stream>>>(ghist, nhist);
    mlp_kernel<<<4, 256, 0, stream>>>(nhist, W1, b1, W2, b2, out);
}